// HungarianLoss_67808943669946
// MI455X (gfx1250) — compile-verified
//
#include <hip/hip_runtime.h>

// D = A(16x4, f32) * B(4x16, f32) + C(16x16, f32)  -- V_WMMA_F32_16X16X4_F32
typedef float v2f __attribute__((ext_vector_type(2)));
typedef float v8f __attribute__((ext_vector_type(8)));

// All 24 permutations of (0,1,2,3), 2 bits per element: perm[p][i] = (kPerm[p]>>(2*i))&3.
__constant__ unsigned char kPerm[24] = {
    0xE4, 0xB4, 0xD8, 0x78, 0x9C, 0x6C,
    0xE1, 0xB1, 0xC9, 0x39, 0x8D, 0x2D,
    0xD2, 0x72, 0xC6, 0x36, 0x4E, 0x1E,
    0x93, 0x63, 0x87, 0x27, 0x4B, 0x1B};

__device__ __forceinline__ float smooth_l1(float d) {  // beta = 1
    return d < 1.0f ? 0.5f * d * d : d - 0.5f;
}

// Core of one 16-box chunk: build smooth-L1 entries (B operand), run the
// transposed permutation matmul D[perm, box] = P x L^T via 8 chained WMMAs,
// return this lane's 24-perm min for box = lane%16 (identical across halves).
__device__ __forceinline__ float chunk_min(const float4 p, const float4 t0, const float4 t1,
                                           const float maskf,
                                           const v2f a0[4], const v2f a1[4], const v8f dpad) {
    const float tx[4] = {t0.x, t0.z, t1.x, t1.z};
    const float ty[4] = {t0.y, t0.w, t1.y, t1.w};

    v2f b[4];
    for (int k = 0; k < 4; ++k) {
        const float lx = smooth_l1(fabsf(p.x - tx[k])) + smooth_l1(fabsf(p.y - ty[k]));
        const float ly = smooth_l1(fabsf(p.z - tx[k])) + smooth_l1(fabsf(p.w - ty[k]));
        b[k].x = lx * maskf;
        b[k].y = ly * maskf;
    }

    v8f d0 = {0.f, 0.f, 0.f, 0.f, 0.f, 0.f, 0.f, 0.f};
    v8f d1 = dpad;  // pad perm rows seeded BIG (lanes 16-31)
    for (int k = 0; k < 4; ++k) {
        d0 = __builtin_amdgcn_wmma_f32_16x16x4_f32(false, a0[k], false, b[k],
                                                   (short)0, d0, false, false);
        d1 = __builtin_amdgcn_wmma_f32_16x16x4_f32(false, a1[k], false, b[k],
                                                   (short)0, d1, false, false);
    }

    // lanes h=0 hold perms {0..7, 16..23}; lanes h=1 hold {8..15, BIG}
    float mn = fminf(d0[0], d0[1]);
    mn = fminf(mn, fminf(d0[2], d0[3]));
    mn = fminf(mn, fminf(d0[4], d0[5]));
    mn = fminf(mn, fminf(d0[6], d0[7]));
    mn = fminf(mn, fminf(d1[0], d1[1]));
    mn = fminf(mn, fminf(d1[2], d1[3]));
    mn = fminf(mn, fminf(d1[4], d1[5]));
    mn = fminf(mn, fminf(d1[6], d1[7]));
    mn = fminf(mn, __shfl_xor(mn, 16, 32));  // combine halves -> full 24-perm min
    return mn;
}

__global__ void hungarian_min_kernel(const float* __restrict__ pred,
                                     const float* __restrict__ target,
                                     const int* __restrict__ weight,
                                     int n, float* __restrict__ partials) {
    const int lane = threadIdx.x & 31;
    const int m    = lane & 15;   // box column (B,D) / perm row (A)
    const int h    = lane >> 4;   // half: selects K-pair {2h,2h+1} of A/B operands
    const int wavesPerBlock = blockDim.x >> 5;
    const int waveId = blockIdx.x * wavesPerBlock + (threadIdx.x >> 5);
    const int nWaves = (int)gridDim.x * wavesPerBlock;

    // Loop-invariant A operands: A[M=perm][K], K = v + 2h per VGPR v.
    // Chunk k covers entries e = 4k + (v+2h) -> i = k, j = v+2h:  A[p][e] = (perm[p][k] == j).
    // Pass 0: perms 0..15.  Pass 1: perms 16..23 in rows 0..7, rows 8..15 zero.
    v2f a0[4], a1[4];
    for (int k = 0; k < 4; ++k) {
        for (int v = 0; v < 2; ++v) {
            const int j = v + 2 * h;
            const float e0 = (((kPerm[m] >> (2 * k)) & 3) == j) ? 1.0f : 0.0f;
            float e1 = 0.0f;
            if (m < 8) e1 = (((kPerm[16 + m] >> (2 * k)) & 3) == j) ? 1.0f : 0.0f;
            if (v == 0) { a0[k].x = e0; a1[k].x = e1; }
            else        { a0[k].y = e0; a1[k].y = e1; }
        }
    }

    // Pass-1 C seed: pad perm rows (lanes 16..31 of every D VGPR) get BIG.
    const float cpad = h ? 3.0e38f : 0.0f;
    v8f dpad;
    for (int i = 0; i < 8; ++i) dpad[i] = cpad;

    const float4* pred4 = (const float4*)pred;
    const float4* tgt4  = (const float4*)target;

    float acc = 0.0f;

    // ---- Main loop: full 16-box chunks, completely guard-free ----
    const int nFull = n >> 4;
    size_t idx = (size_t)waveId * 16 + m;       // this lane's box index
    const size_t step = (size_t)nWaves * 16;
    for (int c = waveId; c < nFull; c += nWaves, idx += step) {
        const float4 p  = pred4[idx * 2 + h];   // lanes 0-15: corners 0,1; 16-31: 2,3
        const float4 t0 = tgt4[idx * 2 + 0];
        const float4 t1 = tgt4[idx * 2 + 1];
        const float maskf = (weight[idx] > 0) ? 1.0f : 0.0f;
        acc += chunk_min(p, t0, t1, maskf, a0, a1, dpad);
    }

    // ---- Tail chunk (n % 16 boxes): wave 0 handles it, fully guarded ----
    if ((n & 15) != 0 && waveId == 0) {
        const int box  = nFull * 16 + m;
        const int boxc = box < n ? box : n - 1;
        const float4 p  = pred4[(size_t)boxc * 2 + h];
        const float4 t0 = tgt4[(size_t)boxc * 2 + 0];
        const float4 t1 = tgt4[(size_t)boxc * 2 + 1];
        const float maskf = (box < n && weight[boxc] > 0) ? 1.0f : 0.0f;
        acc += chunk_min(p, t0, t1, maskf, a0, a1, dpad);
    }

    // Butterfly sum over all 32 lanes counts each box twice -> *0.5 (exact).
    acc += __shfl_xor(acc, 1, 32);
    acc += __shfl_xor(acc, 2, 32);
    acc += __shfl_xor(acc, 4, 32);
    acc += __shfl_xor(acc, 8, 32);
    acc += __shfl_xor(acc, 16, 32);
    const float wavePartial = acc * 0.5f;

    __shared__ float sdata[32];
    if (lane == 0) sdata[threadIdx.x >> 5] = wavePartial;
    __syncthreads();
    if (threadIdx.x == 0) {
        float s = 0.0f;
        for (int w = 0; w < wavesPerBlock; ++w) s += sdata[w];
        partials[blockIdx.x] = s;   // deterministic fixed-order block partial
    }
}

__global__ void hungarian_finalize_kernel(const float* __restrict__ partials, int nparts,
                                          const int* __restrict__ avg_factor,
                                          float* __restrict__ out) {
    if (blockIdx.x == 0 && threadIdx.x == 0) {
        float s = 0.0f;
        for (int i = 0; i < nparts; ++i) s += partials[i];  // fixed order
        out[0] = s * (1.0f / (float)(*avg_factor));         // LOSS_WEIGHT = 1
    }
}

extern "C" void kernel_launch(void* const* d_in, const int* in_sizes, int n_in,
                              void* d_out, int out_size, void* d_ws, size_t ws_size,
                              hipStream_t stream) {
    const float* pred   = (const float*)d_in[0];
    const float* target = (const float*)d_in[1];
    const int*   weight = (const int*)d_in[2];
    const int*   avgf   = (const int*)d_in[3];
    float*       out    = (float*)d_out;
    float*       parts  = (float*)d_ws;

    const int n = in_sizes[2];  // N boxes (weight is [N])

    int blocks = 1024;
    const int maxBlocks = (int)(ws_size / sizeof(float));
    if (maxBlocks > 0 && blocks > maxBlocks) blocks = maxBlocks;
    const int threads = 256;  // 8 wave32 per block

    hungarian_min_kernel<<<blocks, threads, 0, stream>>>(pred, target, weight, n, parts);
    hungarian_finalize_kernel<<<1, 32, 0, stream>>>(parts, blocks, avgf, out);
}